// SpecialSpmm_85847806313254
// MI455X (gfx1250) — compile-verified
//
#include <hip/hip_runtime.h>

typedef __attribute__((ext_vector_type(2))) float v2f;
typedef __attribute__((ext_vector_type(8))) float v8f;

// ---------------------------------------------------------------------------
// Zero-fill d_out (harness poisons it with 0xAA; we must re-init every call).
// ---------------------------------------------------------------------------
__global__ void SpecialSpmm_zero_kernel(float4* __restrict__ out, int n4) {
  int i = blockIdx.x * blockDim.x + threadIdx.x;
  int stride = gridDim.x * blockDim.x;
  for (; i < n4; i += stride) out[i] = make_float4(0.f, 0.f, 0.f, 0.f);
}

// ---------------------------------------------------------------------------
// COO SpMM: out[row[e], :] += values[e] * b[col[e], :],  D = 64.
//
// One wave32 handles 16 edges per iteration. The scale step is computed as
//   tile(16 edges x 16 cols) = diag(v) * G,  G[m,:] = b[col[m], tile]
// via 4 chained V_WMMA_F32_16X16X4_F32 (K=16 as 4 slices of K=4).
// diag rows have one nonzero -> each output element is one exact f32 product.
// All index arithmetic is unsigned 32-bit so the backend emits the GVS
// (SGPR base + 32-bit VGPR offset, scale_offset) addressing form instead of
// sign-extended 64-bit vaddr pairs.
// ---------------------------------------------------------------------------
__global__ __launch_bounds__(256) void SpecialSpmm_85847806313254_kernel(
    const int* __restrict__ idx,     // [2, E]: rows then cols
    const float* __restrict__ vals,  // [E]
    const float* __restrict__ b,     // [N, 64]
    float* __restrict__ out,         // [N, 64]
    int E, int ngroups) {
  const int lane = threadIdx.x & 31;
  const int wid  = (blockIdx.x * blockDim.x + threadIdx.x) >> 5;
  const int nw   = (gridDim.x * blockDim.x) >> 5;

  const unsigned m    = lane & 15;  // edge-in-group mirrored by this lane; N index
  const int      half = lane >> 4;  // lane half selects K pair per ISA striping
  const int      kk   = half << 1;  // A/B: VGPR0 holds K=kk, VGPR1 holds K=kk+1

  for (int g = wid; g < ngroups; g += nw) {
    const int e  = (g << 4) + (int)m;
    const int ee = (e < E) ? e : (E - 1);          // branch-free tail clamp
    int   r = idx[ee];                              // row
    int   c = idx[E + ee];                          // col
    const float w = (e < E) ? vals[ee] : 0.0f;      // zero weight kills tail
    __builtin_assume(r >= 0);
    __builtin_assume(c >= 0);

    // Prefetch next group's edge stream (global_prefetch_b8).
    {
      const int gn = g + nw;
      if (gn < ngroups) {
        const int en = (gn << 4) + (int)m;
        __builtin_prefetch(&idx[en], 0, 0);
        __builtin_prefetch(&idx[E + en], 0, 0);
        __builtin_prefetch(&vals[en], 0, 0);
      }
    }

    // A_j[m,k] = (m == 4j + k) ? v[m] : 0   (16x4 slice of diag(v))
    v2f a[4];
#pragma unroll
    for (int j = 0; j < 4; ++j) {
      const unsigned k0 = 4 * j + kk;
      a[j].x = (m == k0)     ? w : 0.0f;
      a[j].y = (m == k0 + 1) ? w : 0.0f;
    }

    // Columns this lane must gather for each B_j slice (broadcast via shfl).
    unsigned cB0[4], cB1[4];
#pragma unroll
    for (int j = 0; j < 4; ++j) {
      cB0[j] = (unsigned)__shfl(c, 4 * j + kk, 32);
      cB1[j] = (unsigned)__shfl(c, 4 * j + kk + 1, 32);
    }
    // Row ids needed for scatter: C VGPR p holds M = p + 8*half.
    unsigned rM[8];
#pragma unroll
    for (int p = 0; p < 8; ++p)
      rM[p] = (unsigned)__shfl(r, p + (half << 3), 32);

    // Four 16-wide column tiles cover D = 64.
#pragma unroll
    for (int t = 0; t < 4; ++t) {
      const unsigned col = (unsigned)(t * 16) + m;  // N = lane&15
      v8f acc = {};
#pragma unroll
      for (int j = 0; j < 4; ++j) {
        // B_j 4x16 fp32: VGPR0 = row K=kk, VGPR1 = row K=kk+1 (per lane-half).
        v2f bb;
        bb.x = b[(cB0[j] << 6) + col];
        bb.y = b[(cB1[j] << 6) + col];
        acc = __builtin_amdgcn_wmma_f32_16x16x4_f32(
            /*neg_a=*/false, a[j], /*neg_b=*/false, bb,
            /*c_mod=*/(short)0, acc, /*reuse_a=*/false, /*reuse_b=*/false);
      }
      // Scatter-add the 16x16 tile: element (M = p + 8*half, N = m).
#pragma unroll
      for (int p = 0; p < 8; ++p) {
        unsafeAtomicAdd(&out[(rM[p] << 6) + col], acc[p]);
      }
    }
  }
}

// ---------------------------------------------------------------------------
// Launch wrapper.
// d_in: [0]=indices (int32 [2,E]), [1]=values (f32 [E]), [2]=shape scalar,
//       [3]=b (f32 [N,64]).  d_out: f32 [N,64].
// ---------------------------------------------------------------------------
extern "C" void kernel_launch(void* const* d_in, const int* in_sizes, int n_in,
                              void* d_out, int out_size, void* d_ws, size_t ws_size,
                              hipStream_t stream) {
  const int*   idx  = (const int*)d_in[0];
  const float* vals = (const float*)d_in[1];
  const float* b    = (const float*)d_in[3];
  float*       out  = (float*)d_out;

  const int E       = in_sizes[1];          // values element count
  const int ngroups = (E + 15) >> 4;

  // Zero output (out_size = N*64, multiple of 4).
  const int n4 = out_size >> 2;
  int zblocks = (n4 + 255) / 256;
  if (zblocks > 2048) zblocks = 2048;
  SpecialSpmm_zero_kernel<<<zblocks, 256, 0, stream>>>((float4*)out, n4);

  // Main SpMM: grid-stride over 16-edge groups (8 waves per 256-thread block).
  int blocks = (ngroups + 7) / 8;
  if (blocks > 4096) blocks = 4096;
  SpecialSpmm_85847806313254_kernel<<<blocks, 256, 0, stream>>>(
      idx, vals, b, out, E, ngroups);
}